// MPSCalibrator_49048526520726
// MI455X (gfx1250) — compile-verified
//
#include <hip/hip_runtime.h>
#include <hip/hip_bf16.h>

// ---------------------------------------------------------------------------
// MPS log-likelihood on MI455X (gfx1250), fp32 path via V_WMMA_F32_16X16X4_F32
// chi = 24 -> transfer matrices are 576x576. Whole working set is L2-resident
// (~53 MB vs 192 MB L2), so the GEMMs are issue-bound: each wave register-
// blocks a 32x32 output (4 WMMA tiles) so one K-step's 6 VMEM ops feed 4
// v_wmma instructions instead of 3 VMEM per single v_wmma.
// ---------------------------------------------------------------------------

#define CHI      24
#define NMAT     576                 // chi^2
#define N2       (576 * 576)         // elements per 576x576 matrix
#define K_PATHS  16
#define N_SPINS  34

typedef float v2f __attribute__((ext_vector_type(2)));
typedef float v8f __attribute__((ext_vector_type(8)));

// ---------------------------------------------------------------------------
// Build AAu = Au (x) Au, AAd = Ad (x) Ad, E = AAu + AAd, plus transposes
// (transposes give coalesced trace dot-products later).
// ---------------------------------------------------------------------------
__global__ __launch_bounds__(256)
void build_kron(const float* __restrict__ A,
                float* __restrict__ AAu, float* __restrict__ AAd,
                float* __restrict__ E,
                float* __restrict__ AAuT, float* __restrict__ AAdT)
{
    int idx = blockIdx.x * 256 + threadIdx.x;
    if (idx >= N2) return;
    int r = idx / NMAT, c = idx - r * NMAT;
    int i = r / CHI, k = r - i * CHI;
    int j = c / CHI, l = c - j * CHI;
    const float* Au = A;             // A[0], 24x24
    const float* Ad = A + CHI * CHI; // A[1]
    float u = Au[i * CHI + j] * Au[k * CHI + l];
    float d = Ad[i * CHI + j] * Ad[k * CHI + l];
    AAu[idx] = u;
    AAd[idx] = d;
    E[idx]   = u + d;
    AAuT[c * NMAT + r] = u;
    AAdT[c * NMAT + r] = d;
}

// ---------------------------------------------------------------------------
// 576x576x576 fp32 GEMM. One wave per 32x32 output block = 2x2 grid of
// 16x16 WMMA tiles, K stepped by 4 (V_WMMA_F32_16X16X4_F32).
//   D_p = A_p @ B * scale_p
// B is either a fixed matrix (chain GEMMs) or selected per-batch by the
// spin sample (path GEMMs). Spin is uniform over the block -> EXEC stays
// all-ones, as WMMA requires.
//
// Fragment layouts per ISA 7.12.2:
//   A (16x4):  lanes 0-15 hold K={0,1}, lanes 16-31 hold K={2,3}; M = lane&15
//   B (4x16):  same K striping; N = lane&15
//   C/D:       VGPR v -> row v + 8*(lane>>4), col = lane&15
// ---------------------------------------------------------------------------
__global__ __launch_bounds__(32)
void wmma_gemm576(const float* __restrict__ Abase, long strideA,
                  const float* __restrict__ Bfix,
                  const float* __restrict__ Bu, const float* __restrict__ Bd,
                  const int*   __restrict__ spins, int t,
                  const float* __restrict__ scaleVec,
                  float* __restrict__ Dbase, long strideD)
{
    const int p = blockIdx.z;
    const float* Am = Abase + (long)p * strideA;
    const float* B  = Bfix;
    if (spins) B = (spins[p * N_SPINS + t] == 1) ? Bu : Bd;
    float* D = Dbase + (long)p * strideD;

    const int lane = threadIdx.x;        // 0..31
    const int half = lane >> 4;          // 0: K pair {0,1}, 1: K pair {2,3}
    const int l16  = lane & 15;
    const int m0   = blockIdx.y * 32;
    const int n0   = blockIdx.x * 32;

    v8f acc00 = {}, acc01 = {}, acc10 = {}, acc11 = {};

    // A fragment bases: rows m0..m0+15 and m0+16..m0+31
    const float* arow0 = Am + (m0 + l16) * NMAT + 2 * half;
    const float* arow1 = arow0 + 16 * NMAT;
    // B fragment bases: cols n0..n0+15 and n0+16..n0+31
    const float* bcol0 = B + n0 + l16;
    const float* bcol1 = bcol0 + 16;

    for (int kb = 0; kb < NMAT; kb += 4) {
        v2f a0, a1, b0, b1;
        a0.x = arow0[kb];
        a0.y = arow0[kb + 1];
        a1.x = arow1[kb];
        a1.y = arow1[kb + 1];
        int kk = kb + 2 * half;
        b0.x = bcol0[kk * NMAT];
        b0.y = bcol0[(kk + 1) * NMAT];
        b1.x = bcol1[kk * NMAT];
        b1.y = bcol1[(kk + 1) * NMAT];
        // 4 WMMAs reuse the 2 A-frags and 2 B-frags out of registers
        acc00 = __builtin_amdgcn_wmma_f32_16x16x4_f32(false, a0, false, b0,
                                                      (short)0, acc00, false, false);
        acc01 = __builtin_amdgcn_wmma_f32_16x16x4_f32(false, a0, false, b1,
                                                      (short)0, acc01, false, false);
        acc10 = __builtin_amdgcn_wmma_f32_16x16x4_f32(false, a1, false, b0,
                                                      (short)0, acc10, false, false);
        acc11 = __builtin_amdgcn_wmma_f32_16x16x4_f32(false, a1, false, b1,
                                                      (short)0, acc11, false, false);
    }

    float s = scaleVec ? scaleVec[p] : 1.0f;
    // C/D layout: VGPR v -> row (tile_m + v + 8*half), col (tile_n + l16)
    {
        float* dr = D + (m0 + 8 * half) * NMAT + n0 + l16;
#pragma unroll
        for (int v = 0; v < 8; ++v) dr[v * NMAT] = acc00[v] * s;
    }
    {
        float* dr = D + (m0 + 8 * half) * NMAT + n0 + 16 + l16;
#pragma unroll
        for (int v = 0; v < 8; ++v) dr[v * NMAT] = acc01[v] * s;
    }
    {
        float* dr = D + (m0 + 16 + 8 * half) * NMAT + n0 + l16;
#pragma unroll
        for (int v = 0; v < 8; ++v) dr[v * NMAT] = acc10[v] * s;
    }
    {
        float* dr = D + (m0 + 16 + 8 * half) * NMAT + n0 + 16 + l16;
#pragma unroll
        for (int v = 0; v < 8; ++v) dr[v * NMAT] = acc11[v] * s;
    }
}

// ---------------------------------------------------------------------------
// Trace of a 576x576 matrix (chain normalization). Single block, fixed-order.
// ---------------------------------------------------------------------------
__global__ __launch_bounds__(256)
void mat_trace(const float* __restrict__ M, float* __restrict__ tr)
{
    __shared__ float red[256];
    float acc = 0.f;
    for (int i = threadIdx.x; i < NMAT; i += 256)
        acc += M[i * (NMAT + 1)];
    red[threadIdx.x] = acc;
    __syncthreads();
    for (int s = 128; s > 0; s >>= 1) {
        if (threadIdx.x < s) red[threadIdx.x] += red[threadIdx.x + s];
        __syncthreads();
    }
    if (threadIdx.x == 0) tr[0] = red[0];
}

__global__ __launch_bounds__(256)
void mat_scale(float* __restrict__ M, const float* __restrict__ tr)
{
    int i = blockIdx.x * 256 + threadIdx.x;
    if (i < N2) M[i] = M[i] / tr[0];
}

// ---------------------------------------------------------------------------
// Broadcast EN into every path's state buffer.
// ---------------------------------------------------------------------------
__global__ __launch_bounds__(256)
void broadcast_state(const float* __restrict__ EN, float* __restrict__ S)
{
    int i = blockIdx.x * 256 + threadIdx.x;
    int p = blockIdx.y;
    if (i < N2) S[(long)p * N2 + i] = EN[i];
}

// ---------------------------------------------------------------------------
// Stage 1 of tr(S @ AAu), tr(S @ AAd) = <S, AA^T> per path.
// grid = (81, 16 paths, 2 matrices); each block reduces a 4096-elem slab.
// Fixed-order LDS reduction -> deterministic partials.
// ---------------------------------------------------------------------------
__global__ __launch_bounds__(256)
void path_trace_partials(const float* __restrict__ S,
                         const float* __restrict__ AAuT,
                         const float* __restrict__ AAdT,
                         float* __restrict__ partials)
{
    int z = blockIdx.x;                  // 0..80
    int p = blockIdx.y;                  // 0..15
    const float* AA = blockIdx.z ? AAdT : AAuT;
    const float* Sp = S + (long)p * N2;
    int base = z * 4096;
    float acc = 0.f;
    for (int i = threadIdx.x; i < 4096; i += 256)
        acc += Sp[base + i] * AA[base + i];
    __shared__ float red[256];
    red[threadIdx.x] = acc;
    __syncthreads();
    for (int s = 128; s > 0; s >>= 1) {
        if (threadIdx.x < s) red[threadIdx.x] += red[threadIdx.x + s];
        __syncthreads();
    }
    if (threadIdx.x == 0)
        partials[(blockIdx.z * K_PATHS + p) * 81 + z] = red[0];
}

// ---------------------------------------------------------------------------
// Stage 2: fixed-order combine of partials; emit log p and 1/tot per path.
// ---------------------------------------------------------------------------
__global__ __launch_bounds__(16)
void step_combine(const float* __restrict__ partials,
                  const int*   __restrict__ samples, int t,
                  float* __restrict__ inv_tot, float* __restrict__ logp)
{
    int p = threadIdx.x;
    if (p >= K_PATHS) return;
    float tu = 0.f, td = 0.f;
    for (int z = 0; z < 81; ++z) {
        tu += partials[p * 81 + z];
        td += partials[(K_PATHS + p) * 81 + z];
    }
    float tot = tu + td;
    int s = samples[p * N_SPINS + t];
    float psel = ((s == 1) ? tu : td) / tot;
    logp[p * N_SPINS + t] = logf(psel);
    inv_tot[p] = 1.0f / tot;
}

// ---------------------------------------------------------------------------
// Final: out = -sum(logp). Fixed order, single block.
// ---------------------------------------------------------------------------
__global__ __launch_bounds__(256)
void final_reduce(const float* __restrict__ logp, float* __restrict__ out)
{
    __shared__ float red[256];
    float acc = 0.f;
    for (int i = threadIdx.x; i < K_PATHS * N_SPINS; i += 256)
        acc += logp[i];
    red[threadIdx.x] = acc;
    __syncthreads();
    for (int s = 128; s > 0; s >>= 1) {
        if (threadIdx.x < s) red[threadIdx.x] += red[threadIdx.x + s];
        __syncthreads();
    }
    if (threadIdx.x == 0) out[0] = -red[0];
}

// ---------------------------------------------------------------------------
extern "C" void kernel_launch(void* const* d_in, const int* in_sizes, int n_in,
                              void* d_out, int out_size, void* d_ws, size_t ws_size,
                              hipStream_t stream)
{
    (void)in_sizes; (void)n_in; (void)out_size;
    const float* A       = (const float*)d_in[0];   // (2, 24, 24) f32
    const int*   samples = (const int*)d_in[1];     // (16, 34) i32
    float*       out     = (float*)d_out;           // scalar f32

    // ---- workspace layout (floats) ----
    float* w      = (float*)d_ws;
    float* AAu    = w + 0L * N2;
    float* AAd    = w + 1L * N2;
    float* E      = w + 2L * N2;   // later reused as EN
    float* AAuT   = w + 3L * N2;
    float* AAdT   = w + 4L * N2;
    float* E2     = w + 5L * N2;
    float* M0     = w + 6L * N2;
    float* M1     = w + 7L * N2;
    float* tracev   = w + 8L * N2;          // [1]
    float* partials = tracev + 16;          // [2*16*81 = 2592]
    float* inv_tot  = partials + 2592;      // [16]
    float* logp     = inv_tot + 16;         // [544]
    float* S0       = w + 8L * N2 + 4096;   // [16 * N2]
    float* S1       = S0 + (long)K_PATHS * N2;
    size_t need = ((size_t)(8 + 2 * K_PATHS) * N2 + 4096) * sizeof(float);
    if (ws_size < need) return;             // deterministic no-op if ws too small

    const dim3 gemmGrid(NMAT / 32, NMAT / 32, 1);            // 18x18 wave blocks
    const int  kronBlocks = (N2 + 255) / 256;                // 1296

    // 1) Kronecker setup
    build_kron<<<kronBlocks, 256, 0, stream>>>(A, AAu, AAd, E, AAuT, AAdT);

    // 2) E2 = E @ E
    wmma_gemm576<<<gemmGrid, 32, 0, stream>>>(E, 0, E, nullptr, nullptr,
                                              nullptr, 0, nullptr, E2, 0);

    // 3) four trace-normalized squarings: E2k = (E2k @ E2k) / tr
    float* src = E2;
    float* pp[2] = { M0, M1 };
    for (int it = 0; it < 4; ++it) {
        float* dst = pp[it & 1];
        wmma_gemm576<<<gemmGrid, 32, 0, stream>>>(src, 0, src, nullptr, nullptr,
                                                  nullptr, 0, nullptr, dst, 0);
        mat_trace<<<1, 256, 0, stream>>>(dst, tracev);
        mat_scale<<<kronBlocks, 256, 0, stream>>>(dst, tracev);
        src = dst;
    }

    // 4) EN = E2k @ E2   (reuse E buffer as EN)
    float* EN = E;
    wmma_gemm576<<<gemmGrid, 32, 0, stream>>>(src, 0, E2, nullptr, nullptr,
                                              nullptr, 0, nullptr, EN, 0);

    // 5) String_p = EN for all paths
    broadcast_state<<<dim3(kronBlocks, K_PATHS), 256, 0, stream>>>(EN, S0);

    // 6) 34 sequential steps, all 16 paths batched per launch.
    for (int t = 0; t < N_SPINS; ++t) {
        float* Ssrc = (t & 1) ? S1 : S0;
        float* Sdst = (t & 1) ? S0 : S1;
        path_trace_partials<<<dim3(81, K_PATHS, 2), 256, 0, stream>>>(
            Ssrc, AAuT, AAdT, partials);
        step_combine<<<1, 16, 0, stream>>>(partials, samples, t, inv_tot, logp);
        // S_new = (S @ AA_sel) / tot  -- one 576^3 GEMM per path, fused scale
        wmma_gemm576<<<dim3(NMAT / 32, NMAT / 32, K_PATHS), 32, 0, stream>>>(
            Ssrc, (long)N2, nullptr, AAu, AAd, samples, t, inv_tot,
            Sdst, (long)N2);
    }

    // 7) out = -sum(logp)
    final_reduce<<<1, 256, 0, stream>>>(logp, out);
}